// MetadataEncoder_41858751266986
// MI455X (gfx1250) — compile-verified
//
#include <hip/hip_runtime.h>
#include <stdint.h>

typedef __attribute__((ext_vector_type(2))) float v2f;
typedef __attribute__((ext_vector_type(4))) float v4f;
typedef __attribute__((ext_vector_type(8))) float v8f;

#define NUM_TABLES   21
#define NUM_CATS     1000
#define EMB_DIM      16
#define CONCAT_DIM   344          // 21*16 + 8
#define OUT_DIM      32
#define NEG_SLOPE    0.01f

#define ROWS_PER_WAVE   16
#define WAVES_PER_BLOCK 4
#define ROWS_PER_BLOCK  (ROWS_PER_WAVE * WAVES_PER_BLOCK)   // 64
#define BLOCK_THREADS   (WAVES_PER_BLOCK * 32)              // 128

// Padded LDS row stride for the combined tile: 348 floats.
//  - 348*4 = 1392 bytes, multiple of 16 -> b128 async stores aligned
//  - 348 mod 64 = 28, and r*28 mod 64 distinct for r=0..15 -> conflict-free A loads
#define CSTRIDE      348
#define W_LDS_OFF    (ROWS_PER_BLOCK * CSTRIDE)             // 22272 floats
#define LDS_FLOATS   (W_LDS_OFF + OUT_DIM * CONCAT_DIM)     // 33280 floats
// dynamic LDS bytes = 133120 (< 320KB per WGP)

static_assert(CONCAT_DIM % 4 == 0, "K must be multiple of 4 for wmma f32 16x16x4");

__device__ __forceinline__ float lrelu(float v) {
    return v >= 0.f ? v : NEG_SLOPE * v;
}

// One 16-byte async global->LDS transfer (ASYNCcnt tracked, no VGPR data path).
__device__ __forceinline__ void async_ld16(uint32_t lds_addr, const void* gaddr) {
    asm volatile("global_load_async_to_lds_b128 %0, %1, off"
                 :: "v"(lds_addr), "v"(gaddr) : "memory");
}

// 64-byte variant: same address pair, INST_OFFSET applies to BOTH global and LDS
// addresses (CDNA5 ISA), so 4 chunks need only one address setup.
__device__ __forceinline__ void async_ld64(uint32_t lds_addr, const void* gaddr) {
    asm volatile(
        "global_load_async_to_lds_b128 %0, %1, off\n\t"
        "global_load_async_to_lds_b128 %0, %1, off offset:16\n\t"
        "global_load_async_to_lds_b128 %0, %1, off offset:32\n\t"
        "global_load_async_to_lds_b128 %0, %1, off offset:48"
        :: "v"(lds_addr), "v"(gaddr) : "memory");
}

__global__ __launch_bounds__(BLOCK_THREADS, 1)
void metadata_encoder_kernel(const int*   __restrict__ x_cat,    // [B,21]
                             const float* __restrict__ x_cont,   // [B,1]
                             const float* __restrict__ tables,   // [21,1000,16]
                             const float* __restrict__ W_c,      // [8,1]
                             const float* __restrict__ b_c,      // [8]
                             const float* __restrict__ W,        // [32,344] row-major
                             const float* __restrict__ b,        // [32]
                             float*       __restrict__ out,      // [B,32]
                             int batch)
{
    extern __shared__ float smem[];
    const int tid  = threadIdx.x;
    const int lane = tid & 31;
    const int wave = tid >> 5;
    const int wave_row0 = blockIdx.x * ROWS_PER_BLOCK + wave * ROWS_PER_WAVE;

    // Flat LDS byte address of smem: low 32 bits of the flat aperture address
    // are the LDS byte offset (ISA 10.2 aperture mapping).
    const uint32_t lds_base = (uint32_t)(uintptr_t)smem;

    // ---------------- Phase 1: async W -> LDS (fire and forget) -------------
    {
        const char* Wb = (const char*)W;
        const uint32_t wdst = lds_base + (uint32_t)(W_LDS_OFF * 4);
        for (int i = tid; i < (OUT_DIM * CONCAT_DIM * 4) / 16; i += BLOCK_THREADS)
            async_ld16(wdst + (uint32_t)i * 16, Wb + (size_t)i * 16);
    }

    // ---------------- Phase 2: build this wave's combined[16][344] tile -----
    const uint32_t ctile_lds = lds_base + (uint32_t)(wave * ROWS_PER_WAVE * CSTRIDE * 4);
    float* ctile = smem + wave * ROWS_PER_WAVE * CSTRIDE;

    // 2a: embedding gather, 16 rows x 21 tables. One (row,table) pair per
    // lane-step; x_cat reads are fully coalesced (flat = wave_row0*21 + p for
    // in-range rows). Table rows stream straight into LDS via async b128.
    for (int p = lane; p < ROWS_PER_WAVE * NUM_TABLES; p += 32) {
        int r   = p / NUM_TABLES;
        int t   = p - r * NUM_TABLES;
        int row = wave_row0 + r;
        int rc  = row < batch ? row : (batch - 1);     // clamp tail (store guarded)
        int idx = x_cat[rc * NUM_TABLES + t];
        const float* src = tables + ((size_t)t * NUM_CATS + (size_t)idx) * EMB_DIM;
        uint32_t dst = ctile_lds + (uint32_t)(r * CSTRIDE + t * EMB_DIM) * 4;
        async_ld64(dst, src);
    }

    // 2b: continuous branch Linear(1->8) + LeakyReLU, lanes 0..15 own one row.
    // Writes columns 336..343 (disjoint from async writes) via normal ds_store.
    if (lane < ROWS_PER_WAVE) {
        int row = wave_row0 + lane;
        int rc  = row < batch ? row : (batch - 1);
        float xc = x_cont[rc];
        float* dst = ctile + lane * CSTRIDE + NUM_TABLES * EMB_DIM;
        #pragma unroll
        for (int j = 0; j < 8; ++j)
            dst[j] = lrelu(xc * W_c[j] + b_c[j]);
    }

    // All async LDS writes from this wave landed, then block-wide barrier so
    // every wave sees the shared W tile.
    asm volatile("s_wait_asynccnt 0x0" ::: "memory");
    __syncthreads();

    // ---------------- Phase 3: [16,344] x [344,32] via v_wmma_f32_16x16x4_f32
    // A fragment (ISA 7.12.2, 32-bit A 16x4): lanes 0-15 -> K=0,1 ; lanes
    // 16-31 -> K=2,3 ; M = lane&15. B mirrors with N = lane&15.
    const int mrow = lane & 15;              // A: row M ; B/D: column N (in tile)
    const int ksel = (lane >> 4) * 2;        // K sub-offset for this half-wave

    v8f c0, c1;                              // C init = bias broadcast over M
    {
        float bias0 = b[mrow];
        float bias1 = b[16 + mrow];
        #pragma unroll
        for (int v = 0; v < 8; ++v) { c0[v] = bias0; c1[v] = bias1; }
    }

    const float* __restrict__ arow  = ctile + mrow * CSTRIDE + ksel;
    const float* __restrict__ brow0 = smem + W_LDS_OFF + mrow * CONCAT_DIM + ksel;
    const float* __restrict__ brow1 = smem + W_LDS_OFF + (16 + mrow) * CONCAT_DIM + ksel;

    #pragma unroll 2
    for (int k0 = 0; k0 < CONCAT_DIM; k0 += 4) {
        v2f a  = *(const v2f*)(arow  + k0);   // ds_load_b64 (2addr-fused)
        v2f b0 = *(const v2f*)(brow0 + k0);
        v2f b1 = *(const v2f*)(brow1 + k0);
        c0 = __builtin_amdgcn_wmma_f32_16x16x4_f32(false, a, false, b0, (short)0, c0,
                                                   false, false);
        c1 = __builtin_amdgcn_wmma_f32_16x16x4_f32(false, a, false, b1, (short)0, c1,
                                                   false, false);
    }

    // ---------------- Phase 4: LeakyReLU + store -----------------------------
    // D layout: VGPR v, lane l -> M = v + 8*(l>=16), N = l&15.
    const int mbase = (lane >> 4) * 8;
    float* orow = out + (size_t)(wave_row0 + mbase) * OUT_DIM + mrow;

    if (wave_row0 + ROWS_PER_WAVE <= batch) {
        // Fast path: full tile (all but the final block) — no guards, no EXEC churn.
        #pragma unroll
        for (int v = 0; v < 8; ++v) {
            orow[(size_t)v * OUT_DIM]      = lrelu(c0[v]);
            orow[(size_t)v * OUT_DIM + 16] = lrelu(c1[v]);
        }
    } else {
        #pragma unroll
        for (int v = 0; v < 8; ++v) {
            if (wave_row0 + mbase + v < batch) {
                orow[(size_t)v * OUT_DIM]      = lrelu(c0[v]);
                orow[(size_t)v * OUT_DIM + 16] = lrelu(c1[v]);
            }
        }
    }
}

extern "C" void kernel_launch(void* const* d_in, const int* in_sizes, int n_in,
                              void* d_out, int out_size, void* d_ws, size_t ws_size,
                              hipStream_t stream) {
    const int*   x_cat  = (const int*)  d_in[0];
    const float* x_cont = (const float*)d_in[1];
    const float* tables = (const float*)d_in[2];
    const float* W_c    = (const float*)d_in[3];
    const float* b_c    = (const float*)d_in[4];
    const float* W      = (const float*)d_in[5];
    const float* b      = (const float*)d_in[6];
    float*       out    = (float*)d_out;

    const int batch   = in_sizes[0] / NUM_TABLES;     // 500000
    const int nblocks = (batch + ROWS_PER_BLOCK - 1) / ROWS_PER_BLOCK;
    const size_t shmem = (size_t)LDS_FLOATS * sizeof(float);  // 133120 B

    hipLaunchKernelGGL(metadata_encoder_kernel,
                       dim3(nblocks), dim3(BLOCK_THREADS), shmem, stream,
                       x_cat, x_cont, tables, W_c, b_c, W, b, out, batch);
}